// Linear_9174050144275
// MI455X (gfx1250) — compile-verified
//
#include <hip/hip_runtime.h>
#include <hip/hip_bf16.h>
#include <math.h>

// Event forward-warp + bilinear splat (iwe formatting), MI455X / gfx1250.
//
// Reference semantics:
//   loc   = (ev_y, ev_x)                  (integer coords -> gather degenerates to lookup)
//   fx,fy = flow[b, 0/1, y, x]
//   warped = loc + (1-ts) * (fy, fx)
//   norm_ts = 1 - |1 - ts|
//   4-corner bilinear splat of w and w*norm_ts into channel (pol?0:1) and (pol?2:3).
//
// Memory-bound scatter: ~9.6M fp32 atomic adds, all L2-resident (50MB << 192MB L2).
// - 2D grid (gridDim.y = batch): batch index is uniform, no per-thread division.
// - All batch-dependent address math hoisted to scalar (SGPR) base pointers;
//   lanes carry only 32-bit offsets -> SADDR+voffset global ops, no per-lane
//   64-bit pointer arithmetic.
// - Event-stream loads are non-temporal (TH=NT): streamed once, keep L2 for
//   the flow map + output image that the atomics hammer.
// - Non-returning global_atomic_add_f32 (STOREcnt, no waits in the hot path).

static constexpr int H = 480;
static constexpr int W = 640;

__global__ __launch_bounds__(256) void event_warp_splat_kernel(
    const float* __restrict__ flow,   // [B,2,H,W]
    const float* __restrict__ ts,     // [B,N,1]
    const int*   __restrict__ ev_y,   // [B,N]
    const int*   __restrict__ ev_x,   // [B,N]
    const int*   __restrict__ pol,    // [B,N]
    float*       __restrict__ out,    // [B,4,H,W]
    int N)
{
    const int e = blockIdx.x * blockDim.x + threadIdx.x;
    if (e >= N) return;

    // Uniform per-batch bases: pure SALU, shared by the whole wave.
    const int    b     = blockIdx.y;
    const size_t ebase = (size_t)b * (size_t)N;
    const int    plane = H * W;
    const float* fb    = flow + (size_t)b * 2 * (size_t)plane;  // flow[b]
    float*       ob    = out  + (size_t)b * 4 * (size_t)plane;  // out[b]

    // Coalesced, single-use event stream: non-temporal loads (L2 TH=NT),
    // 32-bit lane offset against a scalar base.
    const int   y = __builtin_nontemporal_load(ev_y + ebase + (unsigned)e);
    const int   x = __builtin_nontemporal_load(ev_x + ebase + (unsigned)e);
    const float t = __builtin_nontemporal_load(ts   + ebase + (unsigned)e);
    const int   p = __builtin_nontemporal_load(pol  + ebase + (unsigned)e);

    // Degenerate bilinear gather: integer event coords -> direct lookup (RT, reused).
    const unsigned fo = (unsigned)(y * W + x);
    const float fx = fb[fo];                      // channel 0 = x-flow
    const float fy = fb[fo + (unsigned)plane];    // channel 1 = y-flow

    const float dt = 1.0f - t;
    const float wy = (float)y + dt * fy;
    const float wx = (float)x + dt * fx;
    const float norm_ts = 1.0f - fabsf(1.0f - t);

    const float by = floorf(wy);
    const float bx = floorf(wx);
    const float fry = wy - by;        // in [0,1)
    const float frx = wx - bx;

    // Corner weights (clamp is a no-op for finite inputs since fr in [0,1)).
    const float wgt[4] = { (1.0f - fry) * (1.0f - frx),   // (0,0)
                           (1.0f - fry) * frx,            // (0,1)
                           fry * (1.0f - frx),            // (1,0)
                           fry * frx };                   // (1,1)

    // Channel select folded into a 32-bit offset: pol==1 -> (0,2); pol==0 -> (1,3).
    const unsigned choff = (p == 1) ? 0u : (unsigned)plane;

    #pragma unroll
    for (int c = 0; c < 4; ++c) {
        const float cy = by + (float)(c >> 1);
        const float cx = bx + (float)(c & 1);
        const float w  = wgt[c];
        // In-bounds test on the unclamped float corner (matches reference);
        // skipping exact-zero weights is bit-safe (all contributions >= 0).
        if (cy >= 0.0f && cy <= (float)(H - 1) &&
            cx >= 0.0f && cx <= (float)(W - 1) && w > 0.0f) {
            const unsigned off = choff + (unsigned)((int)cy * W + (int)cx);
            // Non-returning global_atomic_add_f32, SADDR base + 32-bit voffset.
            atomicAdd(ob + off, w);
            atomicAdd(ob + off + 2u * (unsigned)plane, w * norm_ts);
        }
    }
}

extern "C" void kernel_launch(void* const* d_in, const int* in_sizes, int n_in,
                              void* d_out, int out_size, void* d_ws, size_t ws_size,
                              hipStream_t stream) {
    const float* flow = (const float*)d_in[0];   // [B,2,H,W]
    const float* ts   = (const float*)d_in[1];   // [B,N,1]
    const int*   ev_y = (const int*)d_in[2];     // [B,N]
    const int*   ev_x = (const int*)d_in[3];     // [B,N]
    const int*   pol  = (const int*)d_in[4];     // [B,N]
    float*       out  = (float*)d_out;           // [B,4,H,W]

    const int B = in_sizes[0] / (2 * H * W);
    const int N = in_sizes[2] / (B > 0 ? B : 1);

    // d_out is poisoned; scatter-add needs zeros. Memset node is graph-capture legal.
    hipMemsetAsync(out, 0, (size_t)out_size * sizeof(float), stream);

    const int threads = 256;                     // 8 wave32 waves per block
    dim3 grid((N + threads - 1) / threads, B, 1);
    event_warp_splat_kernel<<<grid, threads, 0, stream>>>(
        flow, ts, ev_y, ev_x, pol, out, N);
}